// Encoder3_16054587752729
// MI455X (gfx1250) — compile-verified
//
#include <hip/hip_runtime.h>
#include <hip/hip_bf16.h>

typedef __attribute__((ext_vector_type(2))) float v2f;
typedef __attribute__((ext_vector_type(4))) float v4f;
typedef __attribute__((ext_vector_type(8))) float v8f;

#define N_NODES 50000
#define N_EDGES 600000
#define IN_CH   128
#define HID_CH  256

// ---------------------------------------------------------------------------
// Phase 0: zero the aggregation buffer (d_ws), 50000*128 floats = 25.6 MB.
// ---------------------------------------------------------------------------
__global__ void enc3_zero_agg(float* __restrict__ agg) {
    int i = blockIdx.x * blockDim.x + threadIdx.x;
    const int n4 = (N_NODES * IN_CH) / 4;  // 1.6M float4
    if (i < n4) {
        v4f z = {0.f, 0.f, 0.f, 0.f};
        ((v4f*)agg)[i] = z;
    }
}

// ---------------------------------------------------------------------------
// Phase 1: SpMM scatter-add. One wave (32 lanes) per edge; each lane owns 4
// contiguous channels -> one global_load_b128 of x[col] + 4 f32 atomics.
// x (25.6 MB) is L2-resident, so gathers and atomics stay on-die.
// ---------------------------------------------------------------------------
__global__ void enc3_spmm_scatter(const float* __restrict__ x,
                                  const int*   __restrict__ rows,
                                  const int*   __restrict__ cols,
                                  const float* __restrict__ vals,
                                  float*       __restrict__ agg) {
    int tid  = blockIdx.x * blockDim.x + threadIdx.x;
    int e    = tid >> 5;
    int lane = tid & 31;
    if (e >= N_EDGES) return;

    int   r = rows[e];
    int   c = cols[e];
    float v = vals[e];

    const v4f xv = ((const v4f*)(x + (size_t)c * IN_CH))[lane];
    float* dst = agg + (size_t)r * IN_CH + lane * 4;
    atomicAdd(dst + 0, v * xv.x);
    atomicAdd(dst + 1, v * xv.y);
    atomicAdd(dst + 2, v * xv.z);
    atomicAdd(dst + 3, v * xv.w);
}

// ---------------------------------------------------------------------------
// Phase 2: h = agg @ W + b, PReLU with V_WMMA_F32_16X16X4_F32.
// Each wave computes a 16x64 output strip: one A fragment feeds 4 independent
// accumulators per K-step (A traffic cut 4x vs one-tile-per-wave; the 4 WMMAs
// have no mutual RAW so they pipeline on the matrix unit).
//
// f32 16x4 A layout : lane l holds M = l&15, K = (l>>4)*2 + {0,1}  (2 VGPRs)
// f32 4x16 B layout : lane l holds N = l&15, K = (l>>4)*2 + {0,1}  (2 VGPRs)
// f32 16x16 C/D     : VGPR v, lane l -> M = v + (l>>4)*8, N = l&15 (8 VGPRs)
// ---------------------------------------------------------------------------
__global__ void enc3_gemm_wmma(const float* __restrict__ agg,
                               const float* __restrict__ W,
                               const float* __restrict__ bias,
                               const float* __restrict__ alpha,
                               float*       __restrict__ out) {
    const int nGrps  = HID_CH / 64;                       // 4 strips of 64 cols
    const int tilesM = N_NODES / 16;                      // 3125 (exact)
    int wavesPerBlk  = blockDim.x >> 5;
    int wave  = blockIdx.x * wavesPerBlk + (threadIdx.x >> 5);
    int tileM = wave / nGrps;
    int ngrp  = wave % nGrps;
    if (tileM >= tilesM) return;                          // retires whole waves only

    int lane = threadIdx.x & 31;
    int half = lane >> 4;                                 // 0 or 1
    int l16  = lane & 15;
    int row  = tileM * 16 + l16;                          // A-row this lane feeds
    int col0 = ngrp * 64 + l16;                           // first of 4 B/C/D cols

    const float* Arow = agg + (size_t)row * IN_CH;
    v8f c0 = {}, c1 = {}, c2 = {}, c3 = {};

    #pragma unroll 4
    for (int k = 0; k < IN_CH; k += 4) {
        int kk = k + half * 2;
        v2f a = *(const v2f*)(Arow + kk);                 // contiguous -> b64 load
        const float* Wp = W + (size_t)kk * HID_CH + col0; // row kk, then row kk+1
        v2f b0, b1, b2, b3;
        b0.x = Wp[0];  b0.y = Wp[HID_CH + 0];
        b1.x = Wp[16]; b1.y = Wp[HID_CH + 16];
        b2.x = Wp[32]; b2.y = Wp[HID_CH + 32];
        b3.x = Wp[48]; b3.y = Wp[HID_CH + 48];
        // 8 args: (neg_a, A, neg_b, B, c_mod, C, reuse_a, reuse_b)
        c0 = __builtin_amdgcn_wmma_f32_16x16x4_f32(false, a, false, b0, (short)0, c0, false, false);
        c1 = __builtin_amdgcn_wmma_f32_16x16x4_f32(false, a, false, b1, (short)0, c1, false, false);
        c2 = __builtin_amdgcn_wmma_f32_16x16x4_f32(false, a, false, b2, (short)0, c2, false, false);
        c3 = __builtin_amdgcn_wmma_f32_16x16x4_f32(false, a, false, b3, (short)0, c3, false, false);
    }

    v8f acc[4] = {c0, c1, c2, c3};
    #pragma unroll
    for (int j = 0; j < 4; ++j) {
        int col  = col0 + j * 16;
        float bc = bias[col];
        float al = alpha[col];
        #pragma unroll
        for (int v = 0; v < 8; ++v) {
            float h = acc[j][v] + bc;
            h = (h > 0.f) ? h : al * h;
            int orow = tileM * 16 + v + half * 8;
            out[(size_t)orow * HID_CH + col] = h;
        }
    }
}

// ---------------------------------------------------------------------------
// Inputs (setup_inputs order): x, edge_rows, edge_cols, edge_vals, W, b, alpha
// Output: float[50000 * 256].  d_ws holds agg (25.6 MB).
// ---------------------------------------------------------------------------
extern "C" void kernel_launch(void* const* d_in, const int* in_sizes, int n_in,
                              void* d_out, int out_size, void* d_ws, size_t ws_size,
                              hipStream_t stream) {
    const float* x     = (const float*)d_in[0];
    const int*   rows  = (const int*)  d_in[1];
    const int*   cols  = (const int*)  d_in[2];
    const float* vals  = (const float*)d_in[3];
    const float* W     = (const float*)d_in[4];
    const float* bias  = (const float*)d_in[5];
    const float* alpha = (const float*)d_in[6];
    float*       out   = (float*)d_out;
    float*       agg   = (float*)d_ws;

    // Phase 0: zero agg. 1.6M float4 / 256 threads = 6250 blocks.
    enc3_zero_agg<<<6250, 256, 0, stream>>>(agg);

    // Phase 1: 600000 edges * 32 lanes / 256 threads = 75000 blocks.
    enc3_spmm_scatter<<<75000, 256, 0, stream>>>(x, rows, cols, vals, agg);

    // Phase 2: 3125 M-tiles * 4 N-strips = 12500 waves; 4 waves/block of 128.
    enc3_gemm_wmma<<<3125, 128, 0, stream>>>(agg, W, bias, alpha, out);
}